// RelativePositionEncoding_17351667876416
// MI455X (gfx1250) — compile-verified
//
#include <hip/hip_runtime.h>
#include <cstdint>
#include <cstddef>

// Problem constants from the reference
#define R_MAX   32
#define S_MAX   2
#define C_Z     128
#define N_POS   66              // 2*(R_MAX+1)
#define IN_DIM  139             // 2*N_POS + 1 + 2*(S_MAX+1)
#define NV4     (C_Z / 4)       // 32 float4 per W row
#define BLOCK   256
#define JSPLIT  2

typedef float f4 __attribute__((ext_vector_type(4)));

__device__ __forceinline__ int iclamp(int x, int lo, int hi) {
    x = x < lo ? lo : x;
    return x > hi ? hi : x;
}

// Row map in W (and in the LDS copy):
//   [0..65]    W_pos   (index d_res,   d_res==65 when !same_chain)
//   [66..131]  W_tok   (index d_tok,   d_tok==65 when !(same_chain&&same_res))
//   [132]      w_ent   (added iff same_ent)
//   [133..138] W_ch    (index d_chain, d_chain==5 when !same_ent)
extern "C" __global__ void __launch_bounds__(BLOCK)
relpos_kernel(const int* __restrict__ asym, const int* __restrict__ resi,
              const int* __restrict__ ent,  const int* __restrict__ sym,
              const int* __restrict__ tok,  const float* __restrict__ W,
              float* __restrict__ out, int N)
{
    extern __shared__ float sW[];                 // IN_DIM * C_Z floats = 71168 B
    const int tid  = threadIdx.x;
    const int nvec = (IN_DIM * C_Z) / 4;          // 4448 float4 chunks

    // ---- Stage W into LDS via CDNA5 async global->LDS DMA (ASYNCcnt) ----
#if defined(__AMDGCN__) && defined(__gfx1250__)
    {
        // LDS byte offset of the dynamic shared base (generic -> AS3 cast).
        const unsigned lds_base =
            (unsigned)(size_t)(__attribute__((address_space(3))) void*)sW;
        const f4* gsrc = (const f4*)W;
        for (int k = tid; k < nvec; k += BLOCK) {
            const unsigned loff = lds_base + (unsigned)k * 16u;
            const f4*      gp   = gsrc + k;
            asm volatile("global_load_async_to_lds_b128 %0, %1, off"
                         :: "v"(loff), "v"(gp)
                         : "memory");
        }
        asm volatile("s_wait_asynccnt 0" ::: "memory");
    }
#else
    // Host pass / non-gfx1250 fallback: plain global load + ds store
    for (int k = tid; k < nvec; k += BLOCK)
        ((f4*)sW)[k] = ((const f4*)W)[k];
#endif
    __syncthreads();

    const int i     = blockIdx.x;                 // output row
    const int lane  = tid & 31;                   // channel group (lane*4 .. lane*4+3)
    const int wave  = tid >> 5;                   // 8 waves per block
    const int jspan = N / JSPLIT;
    const int j0    = blockIdx.y * jspan;

    // i-side scalars: block-uniform -> scalar loads
    const int a_i = asym[i], r_i = resi[i], e_i = ent[i], s_i = sym[i], t_i = tok[i];

    const f4* sW4  = (const f4*)sW;
    f4*       out4 = (f4*)out + (size_t)i * (size_t)N * NV4;

    for (int jj = wave; jj < jspan; jj += (BLOCK / 32)) {
        const int j = j0 + jj;
        // j-side scalars: wave-uniform addresses, hot in cache
        const int a_j = asym[j], r_j = resi[j], e_j = ent[j], s_j = sym[j], t_j = tok[j];

        const bool same_chain = (a_i == a_j);
        const bool same_res   = (r_i == r_j);
        const bool same_ent   = (e_i == e_j);

        const int drow = same_chain ? iclamp(r_i - r_j + R_MAX, 0, 2 * R_MAX)
                                    : (2 * R_MAX + 1);
        const int trow = (same_chain && same_res) ? iclamp(t_i - t_j + R_MAX, 0, 2 * R_MAX)
                                                  : (2 * R_MAX + 1);
        const int crow = same_ent ? iclamp(s_i - s_j + S_MAX, 0, 2 * S_MAX)
                                  : (2 * S_MAX + 1);

        // 4x ds_load_b128 gathers from the LDS-resident table
        const f4 vp = sW4[drow * NV4 + lane];
        const f4 vt = sW4[(N_POS + trow) * NV4 + lane];
        const f4 vc = sW4[(2 * N_POS + 1 + crow) * NV4 + lane];
        const f4 ve = sW4[(2 * N_POS) * NV4 + lane];

        const float se = same_ent ? 1.0f : 0.0f;
        f4 r = vp + vt + vc;
        r += se * ve;

        // One coalesced 512B store per wave, non-temporal (output >> L2)
        __builtin_nontemporal_store(r, out4 + (size_t)j * NV4 + lane);
    }
}

extern "C" void kernel_launch(void* const* d_in, const int* in_sizes, int n_in,
                              void* d_out, int out_size, void* d_ws, size_t ws_size,
                              hipStream_t stream) {
    const int*   asym = (const int*)d_in[0];
    const int*   resi = (const int*)d_in[1];
    const int*   ent  = (const int*)d_in[2];
    const int*   sym  = (const int*)d_in[3];
    const int*   tok  = (const int*)d_in[4];
    const float* W    = (const float*)d_in[5];
    float*       out  = (float*)d_out;

    const int N = in_sizes[0];                    // B*N with B==1 -> 1024
    const size_t lds_bytes = (size_t)IN_DIM * C_Z * sizeof(float);   // 71168 B

    // 71 KB dynamic LDS exceeds the 64 KB default cap -> raise it (host-side,
    // deterministic, not a stream op; safe under graph capture).
    (void)hipFuncSetAttribute(reinterpret_cast<const void*>(relpos_kernel),
                              hipFuncAttributeMaxDynamicSharedMemorySize,
                              (int)lds_bytes);

    dim3 grid((unsigned)N, JSPLIT);
    relpos_kernel<<<grid, BLOCK, lds_bytes, stream>>>(asym, resi, ent, sym, tok, W, out, N);
}